// PolyGCLayer_75582834475565
// MI455X (gfx1250) — compile-verified
//
#include <hip/hip_runtime.h>
#include <hip/hip_bf16.h>

typedef float v8f __attribute__((ext_vector_type(8)));
typedef float v2f __attribute__((ext_vector_type(2)));

#define NNODES 10000
#define NEDGES 160000
#define NB     32
#define NF     64
#define NK     5
#define NFO    128
#define FB     2048          // NF * NB
#define KDIM   320           // NF * NK
#define POOL   4
#define NGRP   2500          // NNODES / POOL

// ---------------------------------------------------------------------------
// x [B,N,F] -> x0 [N, F*B] with x0[n, f*32+b] = x[b,n,f]
// ---------------------------------------------------------------------------
__global__ __launch_bounds__(256) void transpose_x_kernel(
    const float* __restrict__ x, float* __restrict__ x0)
{
    __shared__ float tile[FB];
    const int n = blockIdx.x;
    for (int idx = threadIdx.x; idx < FB; idx += 256) {
        int b = idx >> 6;          // 0..31
        int f = idx & 63;          // 0..63
        tile[f * NB + b] = x[((size_t)b * NNODES + n) * NF + f];
    }
    __syncthreads();
    for (int idx = threadIdx.x; idx < FB; idx += 256)
        x0[(size_t)n * FB + idx] = tile[idx];
}

// dst = -src (elementwise, float4)
__global__ __launch_bounds__(256) void negate_copy4_kernel(
    const float4* __restrict__ src, float4* __restrict__ dst, int n4)
{
    int i = blockIdx.x * 256 + threadIdx.x;
    if (i < n4) {
        float4 v = src[i];
        dst[i] = make_float4(-v.x, -v.y, -v.z, -v.w);
    }
}

// ---------------------------------------------------------------------------
// out[row] += scale * val[e] * in[col]  (rows of length 2048, atomic scatter;
// both 82MB matrices are L2-resident on the 192MB L2)
// ---------------------------------------------------------------------------
__global__ __launch_bounds__(512) void spmm_scatter_kernel(
    const int* __restrict__ rows, const int* __restrict__ cols,
    const float* __restrict__ vals, const float* __restrict__ in,
    float* out, float scale)
{
    const int e = blockIdx.x;
    const int r = rows[e];
    const int c = cols[e];
    const float s = vals[e] * scale;
    const int t = threadIdx.x;                       // 0..511
    const float4 v = ((const float4*)(in + (size_t)c * FB))[t];
    float* o = out + (size_t)r * FB + t * 4;
    atomicAdd(o + 0, s * v.x);
    atomicAdd(o + 1, s * v.y);
    atomicAdd(o + 2, s * v.z);
    atomicAdd(o + 3, s * v.w);
}

// ---------------------------------------------------------------------------
// Async stage one node's 5 term rows (5*8KB = 40KB) into an LDS buffer using
// CDNA5 async global->LDS (tracked by ASYNCcnt, bypasses VGPRs).
// Generic LDS pointers truncate to the LDS byte address (ISA 10.2).
// ---------------------------------------------------------------------------
__device__ __forceinline__ void stage_node_async(
    float* dst, const float* __restrict__ terms, int n, int tid)
{
    const char* nodebase = (const char*)(terms + (size_t)n * FB);
#pragma unroll
    for (int it = 0; it < 10; ++it) {                 // 2560 float4 / 256 thr
        const int idx = it * 256 + tid;               // float4 index
        const int k   = idx >> 9;                     // term id (512 f4/term)
        const int r4  = idx & 511;
        const char* src = nodebase + (size_t)k * ((size_t)NNODES * FB * 4)
                                   + (size_t)r4 * 16;
        const unsigned ldsa = (unsigned)(uintptr_t)(dst + idx * 4);
        asm volatile("global_load_async_to_lds_b128 %0, %1, off"
                     :: "v"(ldsa), "v"(src) : "memory");
    }
}

__device__ __forceinline__ void wait_async_all()
{
    asm volatile("s_wait_asynccnt 0" ::: "memory");
}

// ---------------------------------------------------------------------------
// Fused GEMM + bias + ReLU + max-pool(4 nodes), fp32 WMMA 16x16x4.
// One block per pooling group g. K-dim permuted to c' = k*64 + f so that
//   A[b, c'] = term_k[n][f*32+b]  (contiguous row per node, async-staged)
//   W'[c',o] = weight[f*5+k][o]   (permuted once into LDS, reused 4x)
// Both LDS read addresses are linear in the K-step j:
//   A: j*128 + half*64 + m*16 + lm   (pair +32)
//   B: j*512 + half*256 + otile*16 + lm  (pair +128)
// 8 waves: wave w -> M-tile (w&1) over b, N-tiles {2*(w>>1), 2*(w>>1)+1}.
// Double-buffered LDS staging overlaps next node's loads with WMMA.
// ---------------------------------------------------------------------------
__global__ __launch_bounds__(256) void cheb_gemm_relu_pool_kernel(
    const float* __restrict__ terms,   // 5 buffers of [NNODES][2048]
    const float* __restrict__ weight,  // [320][128], row order f*5+k
    const float* __restrict__ bias,    // [128]
    float* __restrict__ out)           // [32][2500][128]
{
    extern __shared__ float smem[];
    float* w_lds  = smem;                      // 320*128 = 160KB
    float* t_lds0 = smem + KDIM * NFO;         // 5*2048  =  40KB
    float* t_lds1 = t_lds0 + NK * FB;          // 5*2048  =  40KB

    const int g   = blockIdx.x;
    const int tid = threadIdx.x;

    // Permuted weight load: w_lds[(k*64+f)*128 + o] = weight[(f*5+k)*128 + o]
    for (int idx = tid; idx < KDIM * NFO; idx += 256) {
        int cp = idx >> 7;
        int o  = idx & 127;
        int k  = cp >> 6;
        int f  = cp & 63;
        w_lds[idx] = weight[(f * NK + k) * NFO + o];
    }

    // Kick off async staging of node 0 while weight stores retire.
    stage_node_async(t_lds0, terms, g * POOL, tid);

    const int lane = tid & 31;
    const int wv   = tid >> 5;
    const int half = lane >> 4;
    const int lm   = lane & 15;
    const int m    = wv & 1;                   // b tile
    const int ot   = (wv >> 1) * 2;            // o tiles ot, ot+1

    const float bias0 = bias[ot * 16 + lm];
    const float bias1 = bias[(ot + 1) * 16 + lm];

    // Constant parts of the linear LDS addresses.
    const int aoff = half * 64 + m * 16 + lm;          // + j*128 (+32 pair)
    const float* bp0 = w_lds + half * 256 + ot * 16 + lm;   // + j*512 (+128)
    const float* bp1 = bp0 + 16;

    v8f r0 = {}, r1 = {};                      // ReLU >= 0 -> 0-init is exact

    for (int node = 0; node < POOL; ++node) {
        wait_async_all();                      // my async fills of current buf
        __syncthreads();                       // everyone's fills + prior reads
        if (node + 1 < POOL)
            stage_node_async((node & 1) ? t_lds0 : t_lds1,
                             terms, g * POOL + node + 1, tid);

        const float* ap = ((node & 1) ? t_lds1 : t_lds0) + aoff;

        v8f acc0 = {}, acc1 = {};
        // Software-pipelined K loop: fetch j+1 fragments before WMMA of j.
        v2f a, b0, b1;
        a.x  = ap[0];    a.y  = ap[32];
        b0.x = bp0[0];   b0.y = bp0[128];
        b1.x = bp1[0];   b1.y = bp1[128];
#pragma unroll 4
        for (int j = 0; j < KDIM / 4 - 1; ++j) {
            v2f na, nb0, nb1;
            const float* ap2 = ap  + (j + 1) * 128;
            const float* bq0 = bp0 + (j + 1) * 512;
            const float* bq1 = bp1 + (j + 1) * 512;
            na.x  = ap2[0];  na.y  = ap2[32];
            nb0.x = bq0[0];  nb0.y = bq0[128];
            nb1.x = bq1[0];  nb1.y = bq1[128];
            acc0 = __builtin_amdgcn_wmma_f32_16x16x4_f32(
                false, a, false, b0, (short)0, acc0, false, false);
            acc1 = __builtin_amdgcn_wmma_f32_16x16x4_f32(
                false, a, false, b1, (short)0, acc1, false, false);
            a = na; b0 = nb0; b1 = nb1;
        }
        acc0 = __builtin_amdgcn_wmma_f32_16x16x4_f32(
            false, a, false, b0, (short)0, acc0, false, false);
        acc1 = __builtin_amdgcn_wmma_f32_16x16x4_f32(
            false, a, false, b1, (short)0, acc1, false, false);

        // bias + ReLU + running max
        for (int v = 0; v < 8; ++v) {
            float u0 = acc0[v] + bias0; u0 = u0 > 0.f ? u0 : 0.f;
            float u1 = acc1[v] + bias1; u1 = u1 > 0.f ? u1 : 0.f;
            r0[v] = u0 > r0[v] ? u0 : r0[v];
            r1[v] = u1 > r1[v] ? u1 : r1[v];
        }
    }

    // C layout: VGPR v, lane L -> M = v + 8*(L>=16), N = L%16
    for (int v = 0; v < 8; ++v) {
        const int b = m * 16 + half * 8 + v;
        out[((size_t)b * NGRP + g) * NFO + ot * 16 + lm]       = r0[v];
        out[((size_t)b * NGRP + g) * NFO + (ot + 1) * 16 + lm] = r1[v];
    }
}

// ---------------------------------------------------------------------------
extern "C" void kernel_launch(void* const* d_in, const int* in_sizes, int n_in,
                              void* d_out, int out_size, void* d_ws, size_t ws_size,
                              hipStream_t stream) {
    const float* x      = (const float*)d_in[0];
    const int*   rows   = (const int*)  d_in[1];
    const int*   cols   = (const int*)  d_in[2];
    const float* vals   = (const float*)d_in[3];
    const float* weight = (const float*)d_in[4];
    const float* bias   = (const float*)d_in[5];
    float*       out    = (float*)d_out;

    const size_t TS = (size_t)NNODES * FB;           // floats per term (81.92MB)
    float* T0 = (float*)d_ws;                        // needs 5*TS*4 = 409.6MB ws
    float* T1 = T0 + TS;
    float* T2 = T0 + 2 * TS;
    float* T3 = T0 + 3 * TS;
    float* T4 = T0 + 4 * TS;

    // T0 = transpose(x)
    transpose_x_kernel<<<NNODES, 256, 0, stream>>>(x, T0);

    // T1 = L @ T0
    hipMemsetAsync(T1, 0, TS * sizeof(float), stream);
    spmm_scatter_kernel<<<NEDGES, 512, 0, stream>>>(rows, cols, vals, T0, T1, 1.0f);

    const int n4 = (int)(TS / 4);
    const int negGrid = (n4 + 255) / 256;

    // T2 = 2*L@T1 - T0
    negate_copy4_kernel<<<negGrid, 256, 0, stream>>>((const float4*)T0, (float4*)T2, n4);
    spmm_scatter_kernel<<<NEDGES, 512, 0, stream>>>(rows, cols, vals, T1, T2, 2.0f);

    // T3 = 2*L@T2 - T1
    negate_copy4_kernel<<<negGrid, 256, 0, stream>>>((const float4*)T1, (float4*)T3, n4);
    spmm_scatter_kernel<<<NEDGES, 512, 0, stream>>>(rows, cols, vals, T2, T3, 2.0f);

    // T4 = 2*L@T3 - T2
    negate_copy4_kernel<<<negGrid, 256, 0, stream>>>((const float4*)T2, (float4*)T4, n4);
    spmm_scatter_kernel<<<NEDGES, 512, 0, stream>>>(rows, cols, vals, T3, T4, 2.0f);

    // Fused WMMA GEMM + bias + ReLU + max-pool (240KB LDS: weight + 2 bufs)
    const size_t shmem = (size_t)(KDIM * NFO + 2 * NK * FB) * sizeof(float);
    cheb_gemm_relu_pool_kernel<<<NGRP, 256, shmem, stream>>>(T0, weight, bias, out);
}